// MaskedNormalize_31172872634755
// MI455X (gfx1250) — compile-verified
//
#include <hip/hip_runtime.h>

// ---------------------------------------------------------------------------
// MaskedNormalize for x[16, 4, 1024, 1024] fp32.
// valid = (x != 0); per-batch masked mean/var (ddof=1) then normalize.
// Memory-bound: ~768 MiB traffic -> ~34us floor at 23.3 TB/s.
// Pass 1 uses V_WMMA_F32_16X16X4_F32 (ones B-matrix) for the 32-lane
// cross-lane reduction of per-lane partial sums (f32 WMMA required: counts
// ~2.9M exceed f16 range). Deterministic: no float atomics anywhere.
// Streaming loops are counted + manually unrolled so multiple b128 loads are
// in flight per wave with no exec-mask loop overhead.
// ---------------------------------------------------------------------------

typedef float f4  __attribute__((ext_vector_type(4)));
typedef float v2f __attribute__((ext_vector_type(2)));
typedef float v8f __attribute__((ext_vector_type(8)));

#define NBATCH 16
#define BPB    256   // blocks per batch in pass 1 (also #partials per batch)
#define TPB    256   // threads per block = 8 waves (wave32)

// Sum a per-lane fp32 value across all 32 lanes using WMMA f32 16x16x4 with a
// ones B matrix.  WMMA #1: A[m][k] holds p(lane) -> D rows R(m)=p(m)+p(m+16)
// replicated across columns.  WMMA #2..#5 feed (d[2i],d[2i+1]) back as A and
// accumulate via C: total = sum_{m=0..15} R(m), broadcast to every lane.
// Requires EXEC all ones (call from uniform control flow only).
__device__ __forceinline__ float wave_sum32_wmma(float p) {
  v2f a;    a[0] = p;   a[1] = 0.0f;
  v2f ones; ones[0] = 1.0f; ones[1] = 1.0f;
  v8f c = {};
  v8f d = __builtin_amdgcn_wmma_f32_16x16x4_f32(false, a, false, ones,
                                                (short)0, c, false, false);
  v8f acc = {};
#pragma unroll
  for (int i = 0; i < 4; ++i) {
    v2f ai; ai[0] = d[2 * i]; ai[1] = d[2 * i + 1];
    acc = __builtin_amdgcn_wmma_f32_16x16x4_f32(false, ai, false, ones,
                                                (short)0, acc, false, false);
  }
  return acc[0];
}

__device__ __forceinline__ void accum(f4 v, float& cnt, float& s1, float& s2) {
#pragma unroll
  for (int e = 0; e < 4; ++e) {
    float u = v[e];
    cnt += (u != 0.0f) ? 1.0f : 0.0f;   // branchless mask count
    s1  += u;                           // zeros contribute nothing to S1/S2
    s2  += u * u;
  }
}

// Pass 1: per-batch partial sums (count, S1, S2). grid = (BPB, NBATCH).
__global__ void __launch_bounds__(TPB)
mnorm_reduce(const float* __restrict__ x, float* __restrict__ partials,
             int per_batch) {
  const int b   = blockIdx.y;
  const int tid = threadIdx.x;
  const f4* xv  = (const f4*)(x + (size_t)b * per_batch);
  const int nvec   = per_batch >> 2;          // float4 count (1,048,576)
  const int stride = BPB * TPB;               // threads covering this batch
  const int iters  = nvec / stride;           // uniform trip count (16)

  // Two accumulator sets to break FP dependency chains.
  float c0 = 0.0f, a0 = 0.0f, q0 = 0.0f;
  float c1 = 0.0f, a1 = 0.0f, q1 = 0.0f;

  int i = blockIdx.x * TPB + tid;
  int k = 0;
  for (; k + 4 <= iters; k += 4) {            // 4 b128 loads in flight
    f4 v0 = xv[i];
    f4 v1 = xv[i + stride];
    f4 v2 = xv[i + 2 * stride];
    f4 v3 = xv[i + 3 * stride];
    i += 4 * stride;
    accum(v0, c0, a0, q0);
    accum(v1, c1, a1, q1);
    accum(v2, c0, a0, q0);
    accum(v3, c1, a1, q1);
  }
  for (; k < iters; ++k) {                    // uniform tail (empty for ref shape)
    f4 v = xv[i];
    i += stride;
    accum(v, c0, a0, q0);
  }
  float cnt = c0 + c1, s1 = a0 + a1, s2 = q0 + q1;

  // 32-lane reduction on the matrix pipe (EXEC all ones here).
  float wc = wave_sum32_wmma(cnt);
  float w1 = wave_sum32_wmma(s1);
  float w2 = wave_sum32_wmma(s2);

  __shared__ float red[8][3];
  const int wave = tid >> 5;
  if ((tid & 31) == 0) { red[wave][0] = wc; red[wave][1] = w1; red[wave][2] = w2; }
  __syncthreads();
  if (tid == 0) {
    float bc = 0.0f, b1 = 0.0f, b2 = 0.0f;
#pragma unroll
    for (int w = 0; w < TPB / 32; ++w) { bc += red[w][0]; b1 += red[w][1]; b2 += red[w][2]; }
    float* p = partials + (size_t)(b * BPB + blockIdx.x) * 3;
    p[0] = bc; p[1] = b1; p[2] = b2;
  }
}

// Pass 2: one block per batch; deterministic double-precision reduction of the
// BPB partial triples, then mean/scale exactly per the reference algebra.
__global__ void __launch_bounds__(TPB)
mnorm_finalize(const float* __restrict__ partials, float* __restrict__ params) {
  const int b = blockIdx.x;
  double c = 0.0, s1 = 0.0, s2 = 0.0;
  for (int i = threadIdx.x; i < BPB; i += TPB) {
    const float* p = partials + (size_t)(b * BPB + i) * 3;
    c += (double)p[0]; s1 += (double)p[1]; s2 += (double)p[2];
  }
  __shared__ double sc[TPB], sa[TPB], sb[TPB];
  sc[threadIdx.x] = c; sa[threadIdx.x] = s1; sb[threadIdx.x] = s2;
  __syncthreads();
  for (int off = TPB / 2; off > 0; off >>= 1) {
    if (threadIdx.x < off) {
      sc[threadIdx.x] += sc[threadIdx.x + off];
      sa[threadIdx.x] += sa[threadIdx.x + off];
      sb[threadIdx.x] += sb[threadIdx.x + off];
    }
    __syncthreads();
  }
  if (threadIdx.x == 0) {
    double n  = sc[0], S1 = sa[0], S2 = sb[0];
    double mean = S1 / n;
    double T    = S1 - mean * n;                 // sum of centered (~0)
    double m2   = T / n;                         // reference's m2 correction
    double C2   = S2 - 2.0 * mean * S1 + mean * mean * n;  // sum v*centered^2
    double var  = (C2 - 2.0 * m2 * T + n * m2 * m2) / (n - 1.0);
    double scale = 1.0 / (sqrt(var) + 1e-5);
    params[b * 2 + 0] = (float)mean;
    params[b * 2 + 1] = (float)scale;
  }
}

// Pass 3: streaming normalize. out = (x!=0) ? (x-mean)*scale : 0.
// Output is write-once -> non-temporal b128 stores to spare L2 capacity
// (x itself stays RT-cached: 192MB L2 gives real reuse between passes).
__global__ void __launch_bounds__(TPB)
mnorm_apply(const float* __restrict__ x, const float* __restrict__ params,
            float* __restrict__ out, int per_batch) {
  const int b = blockIdx.y;
  const float mean  = params[b * 2 + 0];
  const float scale = params[b * 2 + 1];
  const f4* xv = (const f4*)(x   + (size_t)b * per_batch);
  f4*       ov = (f4*)      (out + (size_t)b * per_batch);
  const int nvec   = per_batch >> 2;
  const int stride = gridDim.x * TPB;
  const int iters  = nvec / stride;           // uniform trip count (4)

  int i = blockIdx.x * TPB + threadIdx.x;
  int k = 0;
  for (; k + 2 <= iters; k += 2) {            // 2 loads + 2 NT stores in flight
    f4 va = xv[i];
    f4 vb = xv[i + stride];
    f4 ra, rb;
#pragma unroll
    for (int e = 0; e < 4; ++e) {
      ra[e] = (va[e] != 0.0f) ? (va[e] - mean) * scale : 0.0f;
      rb[e] = (vb[e] != 0.0f) ? (vb[e] - mean) * scale : 0.0f;
    }
    __builtin_nontemporal_store(ra, &ov[i]);
    __builtin_nontemporal_store(rb, &ov[i + stride]);
    i += 2 * stride;
  }
  for (; k < iters; ++k) {                    // uniform tail (empty for ref shape)
    f4 v = xv[i];
    f4 r;
#pragma unroll
    for (int e = 0; e < 4; ++e)
      r[e] = (v[e] != 0.0f) ? (v[e] - mean) * scale : 0.0f;
    __builtin_nontemporal_store(r, &ov[i]);
    i += stride;
  }
}

extern "C" void kernel_launch(void* const* d_in, const int* in_sizes, int n_in,
                              void* d_out, int out_size, void* d_ws, size_t ws_size,
                              hipStream_t stream) {
  (void)n_in; (void)out_size; (void)ws_size;
  const float* x = (const float*)d_in[0];
  float* out = (float*)d_out;
  const int per_batch = in_sizes[0] / NBATCH;      // 4*1024*1024

  // Workspace layout: [NBATCH*BPB*3] partials | [NBATCH*2] params (floats).
  float* partials = (float*)d_ws;
  float* params   = partials + (size_t)NBATCH * BPB * 3;

  dim3 g1(BPB, NBATCH);
  mnorm_reduce<<<g1, TPB, 0, stream>>>(x, partials, per_batch);

  mnorm_finalize<<<NBATCH, TPB, 0, stream>>>(partials, params);

  dim3 g3(1024, NBATCH);
  mnorm_apply<<<g3, TPB, 0, stream>>>(x, params, out, per_batch);
}